// MSDeformableAttention_59914793779611
// MI455X (gfx1250) — compile-verified
//
#include <hip/hip_runtime.h>
#include <hip/hip_bf16.h>
#include <math.h>

typedef float v2f __attribute__((ext_vector_type(2)));
typedef float v8f __attribute__((ext_vector_type(8)));

#define BATCH   16
#define QLEN    300
#define EMBED   256
#define NHEADS  8
#define HEADDIM 32
#define SUMP    16
#define VLEN    8500   // 80*80 + 40*40 + 20*20 + 10*10

// ---------------------------------------------------------------------------
// GEMM: C[M,N] = A[M,256] * W[N,256]^T + bias[N]   (M = 4800, K = 256)
// One 16x16 output tile per wave via V_WMMA_F32_16X16X4_F32 (64 steps of K=4).
// Block = 256 threads = 8 waves -> 16 rows x 128 cols per block.
// ---------------------------------------------------------------------------
__global__ __launch_bounds__(256) void msda_gemm_wmma(
    const float* __restrict__ A, const float* __restrict__ Wt,
    const float* __restrict__ bias, float* __restrict__ C, int N)
{
  // A slab staged in LDS, padded row stride 260 (260 % 64 == 4 -> the 16 rows
  // of an A-fragment read hit 16 distinct bank groups; ds_load_b64 conflict-free)
  __shared__ float As[16 * 260];

  const int m0 = blockIdx.x * 16;
  // cooperative, fully coalesced load of A[16 x 256]
  #pragma unroll
  for (int i = 0; i < 16; ++i) {
    int idx = (int)threadIdx.x + i * 256;
    int row = idx >> 8, col = idx & 255;
    As[row * 260 + col] = A[(size_t)(m0 + row) * 256 + col];
  }
  __syncthreads();

  const int lane = threadIdx.x & 31;
  const int wid  = threadIdx.x >> 5;
  const int n0   = blockIdx.y * 128 + wid * 16;
  if (n0 >= N) return;

  // ISA A-fragment layout (16x4 f32): lane l<16: M=l,    K = k+{0,1}
  //                                   lane l>=16: M=l-16, K = k+{2,3}
  // B (4x16) mirrors it: lane n holds column n%16, same K split.
  const int rsel  = lane & 15;
  const int khalf = (lane >> 4) << 1;   // 0 or 2

  const float* arow = As + rsel * 260 + khalf;                      // LDS
  const float* wrow = Wt + (size_t)(n0 + rsel) * 256 + khalf;       // global (L2-hot)

  v8f acc = {};
  #pragma unroll 8
  for (int k = 0; k < 256; k += 4) {
    float2 a2 = *(const float2*)(arow + k);   // 8B aligned: even indices
    float2 b2 = *(const float2*)(wrow + k);
    v2f a = { a2.x, a2.y };
    v2f b = { b2.x, b2.y };
    acc = __builtin_amdgcn_wmma_f32_16x16x4_f32(
        /*neg_a=*/false, a, /*neg_b=*/false, b,
        /*c_mod=*/(short)0, acc, /*reuse_a=*/false, /*reuse_b=*/false);
  }

  // D layout: VGPR j, lane<16 -> (M=j, N=lane); lane>=16 -> (M=j+8, N=lane-16)
  const int   col   = n0 + rsel;
  const float bb    = bias[col];
  const int   rbase = m0 + ((lane >> 4) << 3);
  #pragma unroll
  for (int j = 0; j < 8; ++j)
    C[(size_t)(rbase + j) * N + col] = acc[j] + bb;
}

// ---------------------------------------------------------------------------
// Phase 2: softmax over 16 points + bilinear sampling + weighted accumulation.
// One wave32 per (b, q, head); lane = channel d in [0,32).
// Every tap load is 32 consecutive floats across lanes -> coalesced 128B.
// ---------------------------------------------------------------------------
__global__ __launch_bounds__(256) void msda_sample_kernel(
    const float* __restrict__ offl,   // [bs*Q, 256] = (head, point, xy)
    const float* __restrict__ attl,   // [bs*Q, 128] = (head, point)
    const float* __restrict__ ref,    // [bs*Q, 4]   = cx, cy, w, h
    const float* __restrict__ value,  // [bs, 8500, 256]
    float* __restrict__ out)          // [bs*Q, 256]
{
  const int lane = threadIdx.x & 31;
  const int w    = blockIdx.x * (blockDim.x >> 5) + (threadIdx.x >> 5);
  if (w >= BATCH * QLEN * NHEADS) return;
  const int head = w & (NHEADS - 1);
  const int bq   = w >> 3;          // b*Q + q
  const int b    = bq / QLEN;

  // per-lane raw offset value: lane d -> point d/2, component d%2
  const float offv = offl[(size_t)bq * 256 + head * 32 + lane];

  // softmax over the 16 attention logits of this head (lanes 0..15)
  float alog = (lane < 16) ? attl[(size_t)bq * 128 + head * 16 + lane] : 0.0f;
  float mx = alog;
  #pragma unroll
  for (int d = 8; d >= 1; d >>= 1) mx = fmaxf(mx, __shfl_xor(mx, d, 32));
  float ex = __expf(alog - mx);
  float sm = ex;
  #pragma unroll
  for (int d = 8; d >= 1; d >>= 1) sm += __shfl_xor(sm, d, 32);
  const float aw = ex / sm;

  const float4 rp = *(const float4*)(ref + (size_t)bq * 4);
  // loc = xy + off * (1/4) * wh * 0.5  ->  scale = wh * 0.125
  const float sx = rp.z * 0.125f;
  const float sy = rp.w * 0.125f;

  const int lvlW[4]  = { 80, 40, 20, 10 };          // square levels
  const int lvlSV[4] = { 0, 6400, 8000, 8400 };

  const float* vbase = value + ((size_t)b * VLEN) * 256 + head * 32 + lane;
  float acc = 0.0f;

  #pragma unroll
  for (int p = 0; p < SUMP; ++p) {
    const int   lvl = p >> 2;
    const int   Wd  = lvlW[lvl];
    const float* lv = vbase + (size_t)lvlSV[lvl] * 256;

    const float ox = __shfl(offv, 2 * p, 32);
    const float oy = __shfl(offv, 2 * p + 1, 32);
    const float wp = __shfl(aw, p, 32);

    // align_corners=False: x_img = loc_x * W - 0.5
    const float x = fmaf(fmaf(ox, sx, rp.x), (float)Wd, -0.5f);
    const float y = fmaf(fmaf(oy, sy, rp.y), (float)Wd, -0.5f);
    const float xf = floorf(x), yf = floorf(y);
    const int   x0 = (int)xf,   y0 = (int)yf;
    const float fx = x - xf,    fy = y - yf;
    const float gx = 1.0f - fx, gy = 1.0f - fy;

    const bool xin0 = (x0 >= 0) && (x0 < Wd);
    const bool xin1 = (x0 + 1 >= 0) && (x0 + 1 < Wd);
    const bool yin0 = (y0 >= 0) && (y0 < Wd);
    const bool yin1 = (y0 + 1 >= 0) && (y0 + 1 < Wd);

    float sample = 0.0f;
    const int rowstride = Wd * 256;
    if (yin0) {
      const float* r0 = lv + (size_t)y0 * rowstride;
      if (xin0) sample = fmaf(gx * gy, r0[(size_t)x0 * 256], sample);
      if (xin1) sample = fmaf(fx * gy, r0[(size_t)(x0 + 1) * 256], sample);
    }
    if (yin1) {
      const float* r1 = lv + (size_t)(y0 + 1) * rowstride;
      if (xin0) sample = fmaf(gx * fy, r1[(size_t)x0 * 256], sample);
      if (xin1) sample = fmaf(fx * fy, r1[(size_t)(x0 + 1) * 256], sample);
    }
    acc = fmaf(wp, sample, acc);
  }

  out[(size_t)bq * 256 + head * 32 + lane] = acc;
}

// ---------------------------------------------------------------------------
extern "C" void kernel_launch(void* const* d_in, const int* in_sizes, int n_in,
                              void* d_out, int out_size, void* d_ws, size_t ws_size,
                              hipStream_t stream)
{
  const float* query = (const float*)d_in[0];   // [16,300,256]
  const float* refp  = (const float*)d_in[1];   // [16,300,1,4]
  const float* value = (const float*)d_in[2];   // [16,8500,256]
  const float* W_off = (const float*)d_in[3];   // [256,256]
  const float* b_off = (const float*)d_in[4];   // [256]
  const float* W_att = (const float*)d_in[5];   // [128,256]
  const float* b_att = (const float*)d_in[6];   // [128]
  // d_in[7] = value_spatial_shapes: compile-time constants, ignored

  float* offl = (float*)d_ws;                          // 4800*256 floats
  float* attl = offl + (size_t)BATCH * QLEN * 256;     // 4800*128 floats

  // off logits: M=4800, N=256  -> grid (300, 2)
  msda_gemm_wmma<<<dim3(QLEN * BATCH / 16, 2), 256, 0, stream>>>(
      query, W_off, b_off, offl, 256);
  // att logits: M=4800, N=128  -> grid (300, 1)
  msda_gemm_wmma<<<dim3(QLEN * BATCH / 16, 1), 256, 0, stream>>>(
      query, W_att, b_att, attl, 128);

  // 38400 waves, 8 waves per block
  msda_sample_kernel<<<(BATCH * QLEN * NHEADS) / 8, 256, 0, stream>>>(
      offl, attl, refp, value, (float*)d_out);
}